// GLMBlock_38517266710912
// MI455X (gfx1250) — compile-verified
//
#include <hip/hip_runtime.h>
#include <hip/hip_bf16.h>
#include <stdint.h>

typedef float v2f __attribute__((ext_vector_type(2)));
typedef float v8f __attribute__((ext_vector_type(8)));

constexpr int B_   = 32;
constexpr int L_   = 4096;
constexpr int H_   = 32;
constexpr int HKV_ = 2;
constexpr int D_   = 128;
constexpr int HID_ = 4096;
constexpr int FFN_ = 13696;
constexpr int QKVN = H_ * D_ + 2 * HKV_ * D_;   // 4608
constexpr int NSPLIT = 16;
constexpr int CHUNK  = L_ / NSPLIT;             // 256
constexpr float EPS_ = 1e-5f;

// ---------------------------------------------------------------------------
// CDNA5 async global->LDS helpers (ASYNCcnt path)
// ---------------------------------------------------------------------------
__device__ __forceinline__ void async_load_b128(uint32_t lds_off, const void* gaddr) {
  asm volatile("global_load_async_to_lds_b128 %0, %1, off"
               :: "v"(lds_off), "v"(gaddr)
               : "memory");
}
__device__ __forceinline__ void wait_async0() {
  asm volatile("s_wait_asynccnt 0x0" ::: "memory");
}

// ---------------------------------------------------------------------------
// RMSNorm: one block per batch row
// ---------------------------------------------------------------------------
__global__ void rmsnorm_kernel(const float* __restrict__ x,
                               const float* __restrict__ w,
                               float* __restrict__ out, int n) {
  const int b = blockIdx.x;
  const float* xr = x + (long)b * n;
  __shared__ float red[256];
  float s = 0.f;
  for (int i = threadIdx.x; i < n; i += blockDim.x) { float v = xr[i]; s += v * v; }
  red[threadIdx.x] = s;
  __syncthreads();
  for (int off = 128; off > 0; off >>= 1) {
    if ((int)threadIdx.x < off) red[threadIdx.x] += red[threadIdx.x + off];
    __syncthreads();
  }
  const float r = rsqrtf(red[0] / (float)n + EPS_);
  for (int i = threadIdx.x; i < n; i += blockDim.x)
    out[(long)b * n + i] = w[i] * xr[i] * r;
}

// ---------------------------------------------------------------------------
// Generic skinny GEMM: C[32 x N] = A[32 x K] @ B[K x N] (+bias) (+residual)
// One wave owns one 16-wide N tile; two 16-row M tiles accumulated via
// V_WMMA_F32_16X16X4_F32 (independent chains -> 2-way WMMA ILP). 4 waves/blk.
// Weights are streamed (zero reuse) so loads go straight to VGPRs; the weight
// stream is prefetched ahead with global_prefetch.
// ---------------------------------------------------------------------------
template <bool HAS_BIAS, bool HAS_RES>
__global__ void gemm32_wmma(const float* __restrict__ A,
                            const float* __restrict__ Bw,
                            const float* __restrict__ bias,
                            const float* __restrict__ res,
                            float* __restrict__ C, int K, int N) {
  const int wave = threadIdx.x >> 5;
  const int lane = threadIdx.x & 31;
  const int lo = lane & 15, hi = lane >> 4;
  const int n0  = (blockIdx.x * 4 + wave) * 16;
  const int col = n0 + lo;

  // A fragment: lane(lo,hi) holds {A[m][k+2hi], A[m][k+2hi+1]}
  const float* a0p = A + (long)lo * K + 2 * hi;          // rows 0..15
  const float* a1p = A + (long)(lo + 16) * K + 2 * hi;   // rows 16..31
  // B fragment: lane(lo,hi) holds {B[k+2hi][col], B[k+2hi+1][col]}
  const float* bp = Bw + col + (long)(2 * hi) * N;

  v8f acc0 = {};
  v8f acc1 = {};
  for (int k = 0; k < K; k += 16) {
#pragma unroll
    for (int kk = 0; kk < 16; kk += 4) {
      v2f a0 = *(const v2f*)(a0p + k + kk);
      v2f a1 = *(const v2f*)(a1p + k + kk);
      v2f b;
      b.x = bp[(long)(k + kk) * N];
      b.y = bp[(long)(k + kk + 1) * N];
      acc0 = __builtin_amdgcn_wmma_f32_16x16x4_f32(false, a0, false, b,
                                                   (short)0, acc0, false, false);
      acc1 = __builtin_amdgcn_wmma_f32_16x16x4_f32(false, a1, false, b,
                                                   (short)0, acc1, false, false);
    }
    if (k + 64 < K) __builtin_prefetch(bp + (long)(k + 64) * N, 0, 1);
  }

  const float bv = HAS_BIAS ? bias[col] : 0.0f;
#pragma unroll
  for (int i = 0; i < 8; ++i) {
    const int r0 = i + 8 * hi;        // rows of acc0
    const int r1 = 16 + i + 8 * hi;   // rows of acc1
    float c0 = acc0[i] + bv;
    float c1 = acc1[i] + bv;
    if (HAS_RES) {
      c0 += res[(long)r0 * N + col];
      c1 += res[(long)r1 * N + col];
    }
    C[(long)r0 * N + col] = c0;
    C[(long)r1 * N + col] = c1;
  }
}

// ---------------------------------------------------------------------------
// RoPE on q/k + extract new k/v rows (caches are NOT mutated).
// grid = (H + HKV, B), block = 64
// ---------------------------------------------------------------------------
__global__ void rope_kernel(const float* __restrict__ qkv,
                            const int* __restrict__ ctx_lens,
                            float* __restrict__ q_rot,
                            float* __restrict__ k_new,
                            float* __restrict__ v_new) {
  const int b = blockIdx.y, h = blockIdx.x, t = threadIdx.x;
  const int pos = ctx_lens[b];
  const float* src;
  float* dst;
  if (h < H_) {
    src = qkv + (long)b * QKVN + (long)h * D_;
    dst = q_rot + ((long)b * H_ + h) * D_;
  } else {
    const int g = h - H_;
    src = qkv + (long)b * QKVN + (long)H_ * D_ + (long)g * D_;
    dst = k_new + ((long)b * HKV_ + g) * D_;
    const float* vs = qkv + (long)b * QKVN + (long)(H_ + HKV_) * D_ + (long)g * D_;
    float* vd = v_new + ((long)b * HKV_ + g) * D_;
    vd[t] = vs[t];
    vd[t + 64] = vs[t + 64];
  }
  if (t < 32) {
    const float inv = __expf(-((float)t / 32.f) * 9.210340371976184f); // 10000^(-t/32)
    const float f = (float)pos * inv;
    const float c = __cosf(f), s = __sinf(f);
    const float x1 = src[2 * t], x2 = src[2 * t + 1];
    dst[2 * t]     = x1 * c - x2 * s;
    dst[2 * t + 1] = x2 * c + x1 * s;
  } else {
    const int d = 64 + (t - 32) * 2;
    dst[d] = src[d];
    dst[d + 1] = src[d + 1];
  }
}

// ---------------------------------------------------------------------------
// Flash-decode attention split kernel. grid = (NSPLIT, HKV, B), block = 32.
// K/V tiles are DMA'd into LDS with GLOBAL_LOAD_ASYNC_TO_LDS_B128 and
// double-pumped against the WMMA work:
//   V[t] streams  ||  S = Q K^T  (LDS-fed WMMAs)
//   K[t+1] streams||  O += P V   (LDS-fed WMMAs)
// ---------------------------------------------------------------------------
__global__ void attn_split_kernel(const float* __restrict__ kc,
                                  const float* __restrict__ vc,
                                  const int* __restrict__ ctx_lens,
                                  const float* __restrict__ q_rot,
                                  const float* __restrict__ k_new,
                                  const float* __restrict__ v_new,
                                  float* __restrict__ part_o,
                                  float* __restrict__ part_m,
                                  float* __restrict__ part_l) {
  const int split = blockIdx.x, g = blockIdx.y, b = blockIdx.z;
  const int lane = threadIdx.x & 31;
  const int lo = lane & 15, hi = lane >> 4;
  const int ctx = ctx_lens[b];
  const long pm_base = (((long)b * HKV_ + g) * NSPLIT + split) * 16;

  const int l_start = split * CHUNK;
  const int l_end = min(l_start + CHUNK, ctx + 1);
  if (l_start > ctx) {   // empty split: mark invalid, skip acc
    if (lane < 16) {
      part_m[pm_base + lane] = -3e38f;
      part_l[pm_base + lane] = 0.f;
    }
    return;
  }

  __shared__ float qs[16 * 128];   // Q tile (pre-scaled)
  __shared__ float kt[16 * 128];   // K tile (async-staged)
  __shared__ float vt[16 * 128];   // V tile (async-staged)
  __shared__ float ps[16 * 16];    // P round-trip for A-fragment layout

  const float scale = 0.08838834764831845f;  // 1/sqrt(128)
  const float* qsrc = q_rot + ((long)b * H_ + (long)g * 16) * D_;
  for (int i = threadIdx.x; i < 16 * 128; i += 32) qs[i] = qsrc[i] * scale;
  __syncthreads();

  const long lstride = (long)HKV_ * D_;
  const float* kb = kc + (long)b * L_ * lstride + (long)g * D_;
  const float* vb = vc + (long)b * L_ * lstride + (long)g * D_;
  const float* kn = k_new + ((long)b * HKV_ + g) * D_;
  const float* vn = v_new + ((long)b * HKV_ + g) * D_;

  const uint32_t kt_lds = (uint32_t)(uintptr_t)&kt[0];
  const uint32_t vt_lds = (uint32_t)(uintptr_t)&vt[0];

  // Each row of a tile = 128 f32 = 32 lanes x 16B: one async b128 per row.
  auto issue_k = [&](int l0) {
#pragma unroll
    for (int j = 0; j < 16; ++j) {
      const int pos = l0 + j;                                // uniform per row
      const float* rp = (pos == ctx) ? kn : (kb + (long)min(pos, L_ - 1) * lstride);
      async_load_b128(kt_lds + (uint32_t)(j * 512 + lane * 16), rp + lane * 4);
    }
  };
  auto issue_v = [&](int l0) {
#pragma unroll
    for (int j = 0; j < 16; ++j) {
      const int pos = l0 + j;
      const float* rp = (pos == ctx) ? vn : (vb + (long)min(pos, L_ - 1) * lstride);
      async_load_b128(vt_lds + (uint32_t)(j * 512 + lane * 16), rp + lane * 4);
    }
  };

  v8f o[8];
  const v8f vzero = {};
#pragma unroll
  for (int n = 0; n < 8; ++n) o[n] = vzero;
  float m_i[8], l_i[8], alpha[8];
#pragma unroll
  for (int i = 0; i < 8; ++i) { m_i[i] = -3e38f; l_i[i] = 0.f; }

  issue_k(l_start);   // prologue: K[0] in flight

  for (int l0 = l_start; l0 < l_end; l0 += 16) {
    wait_async0();    // K[t] resident (prior V writes also retired)
    issue_v(l0);      // V[t] streams while we compute S

    // ---- S = Q @ K^T, two independent WMMA chains ----
    v8f sa = vzero, sb = vzero;
#pragma unroll
    for (int kk = 0; kk < 128; kk += 8) {
      v2f a0 = *(const v2f*)(qs + lo * 128 + kk + 2 * hi);
      v2f b0 = *(const v2f*)(kt + lo * 128 + kk + 2 * hi);
      sa = __builtin_amdgcn_wmma_f32_16x16x4_f32(false, a0, false, b0,
                                                 (short)0, sa, false, false);
      v2f a1 = *(const v2f*)(qs + lo * 128 + kk + 4 + 2 * hi);
      v2f b1 = *(const v2f*)(kt + lo * 128 + kk + 4 + 2 * hi);
      sb = __builtin_amdgcn_wmma_f32_16x16x4_f32(false, a1, false, b1,
                                                 (short)0, sb, false, false);
    }
    v8f s = sa + sb;

    // ---- online softmax (row = i + 8*hi, cols across the 16-lane half) ----
    const int pos = l0 + lo;
    const bool valid = (pos <= ctx);
#pragma unroll
    for (int i = 0; i < 8; ++i) {
      float sv = valid ? s[i] : -3e38f;
      float mx = sv;
      mx = fmaxf(mx, __shfl_xor(mx, 1));
      mx = fmaxf(mx, __shfl_xor(mx, 2));
      mx = fmaxf(mx, __shfl_xor(mx, 4));
      mx = fmaxf(mx, __shfl_xor(mx, 8));
      const float mn = fmaxf(m_i[i], mx);
      const float al = __expf(m_i[i] - mn);
      const float p  = __expf(sv - mn);
      ps[(i + 8 * hi) * 16 + lo] = p;
      float rs = p;
      rs += __shfl_xor(rs, 1);
      rs += __shfl_xor(rs, 2);
      rs += __shfl_xor(rs, 4);
      rs += __shfl_xor(rs, 8);
      l_i[i] = l_i[i] * al + rs;
      m_i[i] = mn;
      alpha[i] = al;
    }
#pragma unroll
    for (int n = 0; n < 8; ++n)
#pragma unroll
      for (int i = 0; i < 8; ++i) o[n][i] *= alpha[i];

    wait_async0();                       // V[t] resident
    if (l0 + 16 < l_end) issue_k(l0 + 16);  // K[t+1] streams during P@V
    __syncthreads();

    // ---- O += P @ V (8 independent chains across d-chunks) ----
#pragma unroll
    for (int kk = 0; kk < 4; ++kk) {
      v2f ap = *(const v2f*)(ps + lo * 16 + kk * 4 + 2 * hi);
      const int r0 = kk * 4 + 2 * hi;    // row within tile
#pragma unroll
      for (int n = 0; n < 8; ++n) {
        v2f bv;
        bv.x = vt[r0 * 128 + n * 16 + lo];
        bv.y = vt[(r0 + 1) * 128 + n * 16 + lo];
        o[n] = __builtin_amdgcn_wmma_f32_16x16x4_f32(false, ap, false, bv,
                                                     (short)0, o[n], false, false);
      }
    }
    __syncthreads();
  }

  // ---- write partials ----
  if (lo == 0) {
#pragma unroll
    for (int i = 0; i < 8; ++i) {
      part_m[pm_base + i + 8 * hi] = m_i[i];
      part_l[pm_base + i + 8 * hi] = l_i[i];
    }
  }
  float* op = part_o + pm_base * D_;
#pragma unroll
  for (int n = 0; n < 8; ++n)
#pragma unroll
    for (int i = 0; i < 8; ++i)
      op[(long)(i + 8 * hi) * D_ + n * 16 + lo] = o[n][i];
}

// ---------------------------------------------------------------------------
// Combine split partials (log-sum-exp merge). grid = (H, B), block = D
// ---------------------------------------------------------------------------
__global__ void attn_combine_kernel(const float* __restrict__ part_o,
                                    const float* __restrict__ part_m,
                                    const float* __restrict__ part_l,
                                    float* __restrict__ ctx_out) {
  const int b = blockIdx.y, h = blockIdx.x;
  const int g = h >> 4, row = h & 15;
  const int d = threadIdx.x;
  const long base = ((long)b * HKV_ + g) * NSPLIT;

  float M = -3e38f;
  for (int s = 0; s < NSPLIT; ++s)
    M = fmaxf(M, part_m[(base + s) * 16 + row]);
  float Lsum = 0.f, val = 0.f;
  for (int s = 0; s < NSPLIT; ++s) {
    const float ls = part_l[(base + s) * 16 + row];
    if (ls > 0.f) {
      const float w = __expf(part_m[(base + s) * 16 + row] - M);
      Lsum += ls * w;
      val += w * part_o[((base + s) * 16 + row) * (long)D_ + d];
    }
  }
  ctx_out[((long)b * H_ + h) * D_ + d] = val / Lsum;
}

// ---------------------------------------------------------------------------
// act = silu(g) * u
// ---------------------------------------------------------------------------
__global__ void silu_mul_kernel(const float* __restrict__ gu,
                                float* __restrict__ act) {
  const long i = (long)blockIdx.x * blockDim.x + threadIdx.x;
  if (i >= (long)B_ * FFN_) return;
  const long b = i / FFN_, j = i % FFN_;
  const float gv = gu[b * (2L * FFN_) + j];
  const float uv = gu[b * (2L * FFN_) + FFN_ + j];
  act[i] = gv / (1.f + __expf(-gv)) * uv;
}

// ---------------------------------------------------------------------------
// Host launcher
// ---------------------------------------------------------------------------
extern "C" void kernel_launch(void* const* d_in, const int* in_sizes, int n_in,
                              void* d_out, int out_size, void* d_ws, size_t ws_size,
                              hipStream_t stream) {
  const float* hidden   = (const float*)d_in[0];
  const int*   ctx_lens = (const int*)d_in[1];
  const float* key_c    = (const float*)d_in[2];
  const float* val_c    = (const float*)d_in[3];
  const float* w_ln1    = (const float*)d_in[4];
  const float* w_ln2    = (const float*)d_in[5];
  const float* w_qkv    = (const float*)d_in[6];
  const float* b_qkv    = (const float*)d_in[7];
  const float* w_dense  = (const float*)d_in[8];
  const float* w_h4h    = (const float*)d_in[9];
  const float* w_4hh    = (const float*)d_in[10];
  float* out = (float*)d_out;

  float* ws = (float*)d_ws;
  float* ln1     = ws; ws += (long)B_ * HID_;
  float* qkv     = ws; ws += (long)B_ * QKVN;
  float* q_rot   = ws; ws += (long)B_ * H_ * D_;
  float* k_new   = ws; ws += (long)B_ * HKV_ * D_;
  float* v_new   = ws; ws += (long)B_ * HKV_ * D_;
  float* part_o  = ws; ws += (long)B_ * HKV_ * NSPLIT * 16 * D_;
  float* part_m  = ws; ws += (long)B_ * HKV_ * NSPLIT * 16;
  float* part_l  = ws; ws += (long)B_ * HKV_ * NSPLIT * 16;
  float* ctx_out = ws; ws += (long)B_ * H_ * D_;
  float* x_res   = ws; ws += (long)B_ * HID_;
  float* ln2     = ws; ws += (long)B_ * HID_;
  float* gu      = ws; ws += (long)B_ * 2 * FFN_;
  float* act     = ws; ws += (long)B_ * FFN_;

  // 1. ln1 = rmsnorm(hidden, w_ln1)
  rmsnorm_kernel<<<B_, 256, 0, stream>>>(hidden, w_ln1, ln1, HID_);
  // 2. qkv = ln1 @ w_qkv + b_qkv
  gemm32_wmma<true, false><<<QKVN / 64, 128, 0, stream>>>(
      ln1, w_qkv, b_qkv, nullptr, qkv, HID_, QKVN);
  // 3. RoPE + new k/v extraction
  rope_kernel<<<dim3(H_ + HKV_, B_), 64, 0, stream>>>(qkv, ctx_lens, q_rot, k_new, v_new);
  // 4. flash-decode attention splits (async K/V staging + WMMA)
  attn_split_kernel<<<dim3(NSPLIT, HKV_, B_), 32, 0, stream>>>(
      key_c, val_c, ctx_lens, q_rot, k_new, v_new, part_o, part_m, part_l);
  // 5. combine splits
  attn_combine_kernel<<<dim3(H_, B_), D_, 0, stream>>>(part_o, part_m, part_l, ctx_out);
  // 6. x = hidden + ctx_out @ w_dense
  gemm32_wmma<false, true><<<HID_ / 64, 128, 0, stream>>>(
      ctx_out, w_dense, nullptr, hidden, x_res, H_ * D_, HID_);
  // 7. ln2 = rmsnorm(x, w_ln2)
  rmsnorm_kernel<<<B_, 256, 0, stream>>>(x_res, w_ln2, ln2, HID_);
  // 8. gu = ln2 @ w_h4h
  gemm32_wmma<false, false><<<(2 * FFN_) / 64, 128, 0, stream>>>(
      ln2, w_h4h, nullptr, nullptr, gu, HID_, 2 * FFN_);
  // 9. act = silu(g) * u
  silu_mul_kernel<<<((long)B_ * FFN_ + 255) / 256, 256, 0, stream>>>(gu, act);
  // 10. out = act @ w_4hh + x
  gemm32_wmma<false, true><<<HID_ / 64, 128, 0, stream>>>(
      act, w_4hh, nullptr, x_res, out, FFN_, HID_);
}